// ContrastiveLoss_28845000360329
// MI455X (gfx1250) — compile-verified
//
#include <hip/hip_runtime.h>
#include <math.h>

// ---------------------------------------------------------------------------
// SCAN-style contrastive loss, fused for MI455X (gfx1250, wave32, WMMA).
//
// Pipeline:
//  k1: SS[t]   = s[t] * s[t]^T          (per-caption 50x50 Gram, f32 WMMA)
//  k2: imn[b,p]= ||im[b,p,:]||          (wave-per-row reduction)
//  k3: per (t,b): G = s[t]*im[b]^T via V_WMMA_F32_16X16X4_F32, then
//      leaky/l2norm/softmax/threshold in LDS; cosine via
//      w12 = W.G and |wc|^2 = W*SS*W^T (Gram trick, no wc materialized);
//      rank-select top-24 of 36, mean -> scores[t,b]
//  k4: hinge loss with max-violation -> scalar
// ---------------------------------------------------------------------------

typedef float v2f __attribute__((ext_vector_type(2)));
typedef float v8f __attribute__((ext_vector_type(8)));

#define B_ 64
#define P_ 36
#define D_ 1024
#define L_ 50
#define LAMBDA_ 9.0f
#define MARGIN_ 0.2f
#define EPS_ 1e-8f

__device__ __forceinline__ v8f wmma_f32(v2f a, v2f b, v8f c) {
  // 8 args: (neg_a, A, neg_b, B, c_mod, C, reuse_a, reuse_b)
  return __builtin_amdgcn_wmma_f32_16x16x4_f32(false, a, false, b, (short)0, c,
                                               false, false);
}

// ---------------- k1: per-caption Gram matrix SS[t] = s[t] s[t]^T ----------
__global__ __launch_bounds__(128) void ss_kernel(const float* __restrict__ s,
                                                 float* __restrict__ SS) {
  const int t    = blockIdx.x;
  const int wave = threadIdx.x >> 5;
  const int lane = threadIdx.x & 31;
  const int half = lane >> 4;
  const int l16  = lane & 15;
  const int koff = half * 2;

  int arow = wave * 16 + l16; if (arow > L_ - 1) arow = L_ - 1;
  const float* sT = s + (size_t)t * L_ * D_;
  const float* ap = sT + (size_t)arow * D_;
  const float* bp[4];
#pragma unroll
  for (int nb = 0; nb < 4; ++nb) {
    int nc = nb * 16 + l16; if (nc > L_ - 1) nc = L_ - 1;
    bp[nb] = sT + (size_t)nc * D_;
  }
  v8f acc[4] = {};
#pragma unroll 2
  for (int k = 0; k < D_; k += 4) {
    v2f a = *reinterpret_cast<const v2f*>(ap + k + koff);
#pragma unroll
    for (int nb = 0; nb < 4; ++nb) {
      v2f bb = *reinterpret_cast<const v2f*>(bp[nb] + k + koff);
      acc[nb] = wmma_f32(a, bb, acc[nb]);
    }
  }
  float* out = SS + (size_t)t * 64 * 64;  // padded 64x64 per t
#pragma unroll
  for (int nb = 0; nb < 4; ++nb)
#pragma unroll
    for (int j = 0; j < 8; ++j) {
      int row = wave * 16 + j + half * 8;
      int col = nb * 16 + l16;
      out[row * 64 + col] = acc[nb][j];
    }
}

// ---------------- k2: image region norms -----------------------------------
__global__ __launch_bounds__(128) void imnorm_kernel(const float* __restrict__ im,
                                                     float* __restrict__ imn) {
  const int b    = blockIdx.x;
  const int wave = threadIdx.x >> 5;
  const int lane = threadIdx.x & 31;
  for (int p = wave; p < P_; p += 4) {
    const float* r = im + ((size_t)b * P_ + p) * D_;
    float sum = 0.f;
    for (int k = lane; k < D_; k += 32) { float v = r[k]; sum += v * v; }
#pragma unroll
    for (int off = 16; off > 0; off >>= 1) sum += __shfl_down(sum, off, 32);
    if (lane == 0) imn[b * P_ + p] = sqrtf(sum);
  }
}

// ---------------- k3: fused attention + cosine score per (t,b) -------------
__global__ __launch_bounds__(128) void score_kernel(
    const float* __restrict__ im, const float* __restrict__ s,
    const int* __restrict__ cap, const float* __restrict__ SS,
    const float* __restrict__ imn, float* __restrict__ scores) {
  const int b = blockIdx.x;
  const int t = blockIdx.y;

  __shared__ float G[L_][40];      // raw dot products  G[l][p]
  __shared__ float A1[L_][40];     // leaky + l2norm'd  A1[l][p]
  __shared__ float SSl[L_][L_ + 1];
  __shared__ float Wm[P_][L_];     // attention weights W[p][l]
  __shared__ float cosv[P_];
  __shared__ float selv[P_];

  const int wave = threadIdx.x >> 5;
  const int lane = threadIdx.x & 31;
  const int half = lane >> 4;
  const int l16  = lane & 15;
  const int koff = half * 2;

  // ---- GEMM1: G[l][p] = <s[t,l,:], im[b,p,:]> via f32 WMMA 16x16x4 ----
  int arow = wave * 16 + l16; if (arow > L_ - 1) arow = L_ - 1;
  const float* ap = s + ((size_t)t * L_ + arow) * D_;
  const float* bp[3];
#pragma unroll
  for (int nb = 0; nb < 3; ++nb) {
    int pc = nb * 16 + l16; if (pc > P_ - 1) pc = P_ - 1;
    bp[nb] = im + ((size_t)b * P_ + pc) * D_;
  }
  v8f acc[3] = {};
#pragma unroll 2
  for (int k = 0; k < D_; k += 4) {
    v2f a = *reinterpret_cast<const v2f*>(ap + k + koff);
#pragma unroll
    for (int nb = 0; nb < 3; ++nb) {
      v2f bb = *reinterpret_cast<const v2f*>(bp[nb] + k + koff);
      acc[nb] = wmma_f32(a, bb, acc[nb]);
    }
  }
#pragma unroll
  for (int nb = 0; nb < 3; ++nb) {
    int col = nb * 16 + l16;
    if (col < P_) {
#pragma unroll
      for (int j = 0; j < 8; ++j) {
        int row = wave * 16 + j + half * 8;
        if (row < L_) G[row][col] = acc[nb][j];
      }
    }
  }
  // Stage SS[t] (50x50) into LDS while GEMM results settle.
  for (int idx = threadIdx.x; idx < L_ * L_; idx += blockDim.x) {
    int r = idx / L_, c = idx - r * L_;
    SSl[r][c] = SS[(size_t)t * 64 * 64 + r * 64 + c];
  }
  __syncthreads();

  // ---- LeakyReLU + L2 norm over regions (per word row l) ----
  for (int l = threadIdx.x; l < L_; l += blockDim.x) {
    float tmp[P_];
    float ss2 = 0.f;
#pragma unroll
    for (int p = 0; p < P_; ++p) {
      float v = G[l][p];
      v = (v >= 0.f) ? v : 0.1f * v;
      tmp[p] = v;
      ss2 += v * v;
    }
    float inv = 1.0f / (sqrtf(ss2) + EPS_);
#pragma unroll
    for (int p = 0; p < P_; ++p) A1[l][p] = tmp[p] * inv;
  }
  __syncthreads();

  int cl = cap[t];
  if (cl < 1) cl = 1;
  if (cl > L_) cl = L_;
  const float inv_cl = 1.0f / (float)cl;

  // ---- per-region softmax over words + threshold + cosine ----
  if (threadIdx.x < P_) {
    const int p = threadIdx.x;
    float mx = -3.0e38f;
    for (int l = 0; l < cl; ++l) mx = fmaxf(mx, A1[l][p] * LAMBDA_);
    float sum = 0.f;
    for (int l = 0; l < cl; ++l) {
      float e = __expf(A1[l][p] * LAMBDA_ - mx);
      Wm[p][l] = e;
      sum += e;
    }
    const float isum = 1.0f / sum;
    float w12 = 0.f;
    for (int l = 0; l < cl; ++l) {
      float w = Wm[p][l] * isum;
      w = (w > inv_cl) ? w : 0.f;   // sparsify
      Wm[p][l] = w;
      w12 += w * G[l][p];           // <im_p, wc_p> via raw dots
    }
    // |wc_p|^2 = W_p . SS . W_p^T  (Gram trick, no D-dim pass)
    float wcn2 = 0.f;
    for (int lp = 0; lp < cl; ++lp) {
      float a_ = 0.f;
      for (int l = 0; l < cl; ++l) a_ += Wm[p][l] * SSl[l][lp];
      wcn2 += a_ * Wm[p][lp];
    }
    const float w1 = imn[b * P_ + p];
    const float w2 = sqrtf(fmaxf(wcn2, 0.f));
    cosv[p] = w12 / fmaxf(w1 * w2, EPS_);
  }
  __syncthreads();

  // ---- drop worst third (rank select, tie-break by index) ----
  if (threadIdx.x < P_) {
    const int p = threadIdx.x;
    const float c = cosv[p];
    int r = 0;
    for (int q = 0; q < P_; ++q) {
      float cq = cosv[q];
      r += (cq < c) || (cq == c && q < p);
    }
    selv[p] = (r >= P_ / 3) ? c : 0.f;
  }
  __syncthreads();
  if (threadIdx.x == 0) {
    float sm = 0.f;
    for (int p = 0; p < P_; ++p) sm += selv[p];
    scores[t * B_ + b] = sm / (float)(P_ - P_ / 3);
  }
}

// ---------------- k4: hinge loss with max-violation ------------------------
__global__ __launch_bounds__(64) void loss_kernel(const float* __restrict__ scores,
                                                  float* __restrict__ out) {
  __shared__ float Sm[64][65];
  __shared__ float dg[64];
  __shared__ float part[64];
  const int i = threadIdx.x;
  for (int j = 0; j < 64; ++j) Sm[i][j] = scores[i * 64 + j];
  dg[i] = scores[i * 64 + i];
  __syncthreads();
  float ms = 0.f, mi = 0.f;
  for (int j = 0; j < 64; ++j) {
    if (j != i) {
      ms = fmaxf(ms, fmaxf(MARGIN_ + Sm[i][j] - dg[i], 0.f));  // cost_s row i
      mi = fmaxf(mi, fmaxf(MARGIN_ + Sm[j][i] - dg[i], 0.f));  // cost_im col i
    }
  }
  part[i] = ms + mi;
  __syncthreads();
  if (i == 0) {
    float tot = 0.f;
    for (int k = 0; k < 64; ++k) tot += part[k];
    out[0] = tot;
  }
}

// ---------------------------------------------------------------------------
extern "C" void kernel_launch(void* const* d_in, const int* in_sizes, int n_in,
                              void* d_out, int out_size, void* d_ws, size_t ws_size,
                              hipStream_t stream) {
  const float* im  = (const float*)d_in[0];  // (64, 36, 1024)
  const float* s   = (const float*)d_in[1];  // (64, 50, 1024)
  const int*   cap = (const int*)d_in[2];    // (64,)

  float* ws     = (float*)d_ws;
  float* SS     = ws;                         // 64 * 64 * 64 floats (padded Gram)
  float* imn    = SS + 64 * 64 * 64;          // 64 * 36
  float* scores = imn + 64 * P_;              // 64 * 64

  ss_kernel<<<64, 128, 0, stream>>>(s, SS);
  imnorm_kernel<<<64, 128, 0, stream>>>(im, imn);
  score_kernel<<<dim3(64, 64), 128, 0, stream>>>(im, s, cap, SS, imn, scores);
  loss_kernel<<<1, 64, 0, stream>>>(scores, (float*)d_out);
}